// HybridBlock_31035433681370
// MI455X (gfx1250) — compile-verified
//
#include <hip/hip_runtime.h>
#include <stdint.h>

// Problem constants (match reference)
namespace {
constexpr int Bb = 2, S = 2048, Hh = 8, Dh = 64, Nn = 32;
constexpr int CH = 32;          // chunks along sequence
constexpr int Lc = S / CH;      // 64 steps per chunk
constexpr float LOG2E = 1.44269504088896340736f;

__device__ __forceinline__ float fast_exp2(float x) {
#if __has_builtin(__builtin_amdgcn_exp2f)
  return __builtin_amdgcn_exp2f(x);   // v_exp_f32
#else
  return exp2f(x);
#endif
}

// 32-bit LDS byte address (generic LDS pointer low 32 bits == DS offset)
__device__ __forceinline__ unsigned lds_addr32(const void* p) {
  return (unsigned)(uintptr_t)p;
}

// CDNA5 async copy: 16B global -> LDS, tracked by ASYNCcnt
__device__ __forceinline__ void async_ld_b128(unsigned lds, const float* g) {
  asm volatile("global_load_async_to_lds_b128 %0, %1, off"
               :: "v"(lds), "v"(g) : "memory");
}
__device__ __forceinline__ void wait_async_then_barrier() {
#if __has_builtin(__builtin_amdgcn_s_wait_asynccnt)
  __builtin_amdgcn_s_wait_asynccnt(0);
#else
  asm volatile("s_wait_asynccnt 0x0" ::: "memory");
#endif
  __syncthreads();
}
} // namespace

// ---------------------------------------------------------------------------
// Kernel 1: per-(b,h,chunk) local scan; emits chunk aggregate (P, Hc) per (d,n)
// layout: agg[((bh*CH + c)*Nn + n)*Dh + d] = {prod dA, local end state}
// ---------------------------------------------------------------------------
__global__ void __launch_bounds__(Dh) ssm_k1_local(
    const float* __restrict__ u, const float* __restrict__ dt,
    const float* __restrict__ A, const float* __restrict__ Bm,
    float2* __restrict__ agg) {
  __shared__ __align__(16) float dt_t[Lc * Dh];
  __shared__ __align__(16) float u_t[Lc * Dh];
  __shared__ __align__(16) float B_t[Lc * Nn];

  const int blk = blockIdx.x;            // bh*CH + c
  const int c = blk % CH, bh = blk / CH;
  const int b = bh / Hh, h = bh % Hh;
  const int t0 = c * Lc;
  const int d = threadIdx.x;             // 0..63

  const float* dt_g = dt + ((size_t)(b * S + t0) * Hh + h) * Dh;
  const float* u_g  = u  + ((size_t)(b * S + t0) * Hh + h) * Dh;
  const float* B_g  = Bm + ((size_t)(b * S + t0) * Hh + h) * Nn;

  // Stage chunk tiles into LDS via the async copy engine (ASYNCcnt path)
  for (int k = d; k < Lc * (Dh / 4); k += Dh) {          // 16 iters/thread
    int row = k / (Dh / 4), col4 = (k % (Dh / 4)) * 4;
    async_ld_b128(lds_addr32(&dt_t[row * Dh + col4]), dt_g + (size_t)row * (Hh * Dh) + col4);
    async_ld_b128(lds_addr32(&u_t [row * Dh + col4]), u_g  + (size_t)row * (Hh * Dh) + col4);
  }
  for (int k = d; k < Lc * (Nn / 4); k += Dh) {          // 8 iters/thread
    int row = k / (Nn / 4), col4 = (k % (Nn / 4)) * 4;
    async_ld_b128(lds_addr32(&B_t[row * Nn + col4]), B_g + (size_t)row * (Hh * Nn) + col4);
  }

  // Overlap the per-lane A-row load with the async copies (before the wait)
  float Ar[Nn];
  const float* A_g = A + ((size_t)h * Dh + d) * Nn;
#pragma unroll
  for (int n = 0; n < Nn; ++n) Ar[n] = A_g[n];

  wait_async_then_barrier();

  float hs[Nn], P[Nn];
#pragma unroll
  for (int n = 0; n < Nn; ++n) { hs[n] = 0.f; P[n] = 1.f; }

  for (int t = 0; t < Lc; ++t) {
    const float dtv = dt_t[t * Dh + d];
    const float uv  = u_t[t * Dh + d];
    const float sc  = -dtv * LOG2E;
    const float dtu = dtv * uv;
#pragma unroll
    for (int n = 0; n < Nn; ++n) {
      const float a = fast_exp2(sc * Ar[n]);             // exp(-dt*A)
      hs[n] = __builtin_fmaf(a, hs[n], B_t[t * Nn + n] * dtu);
      P[n] *= a;
    }
  }

  float2* outp = agg + (size_t)blk * (Nn * Dh);
#pragma unroll
  for (int n = 0; n < Nn; ++n) outp[(size_t)n * Dh + d] = make_float2(P[n], hs[n]);
}

// ---------------------------------------------------------------------------
// Kernel 2: inter-chunk scan. One thread per (bh, n, d); writes each chunk's
// incoming state h0[((bh*CH + c)*Nn + n)*Dh + d]. Fully coalesced per c-step.
// ---------------------------------------------------------------------------
__global__ void __launch_bounds__(256) ssm_k2_scan(
    const float2* __restrict__ agg, float* __restrict__ h0) {
  const int tid = blockIdx.x * blockDim.x + threadIdx.x;  // bh*(Nn*Dh) + n*Dh + d
  if (tid >= Bb * Hh * Nn * Dh) return;
  const int bh = tid / (Nn * Dh);
  const int nd = tid % (Nn * Dh);

  float acc = 0.f;                                        // state entering chunk 0
  for (int c = 0; c < CH; ++c) {
    const size_t idx = (size_t)(bh * CH + c) * (Nn * Dh) + nd;
    const float2 ph = agg[idx];
    h0[idx] = acc;
    acc = __builtin_fmaf(ph.x, acc, ph.y);
  }
}

// ---------------------------------------------------------------------------
// Kernel 3: replay each chunk from its incoming state, emit y
// ---------------------------------------------------------------------------
__global__ void __launch_bounds__(Dh) ssm_k3_out(
    const float* __restrict__ u, const float* __restrict__ dt,
    const float* __restrict__ A, const float* __restrict__ Bm,
    const float* __restrict__ Cm, const float* __restrict__ Dv,
    const float* __restrict__ h0, float* __restrict__ out) {
  __shared__ __align__(16) float dt_t[Lc * Dh];
  __shared__ __align__(16) float u_t[Lc * Dh];
  __shared__ __align__(16) float B_t[Lc * Nn];
  __shared__ __align__(16) float C_t[Lc * Nn];

  const int blk = blockIdx.x;
  const int c = blk % CH, bh = blk / CH;
  const int b = bh / Hh, h = bh % Hh;
  const int t0 = c * Lc;
  const int d = threadIdx.x;

  const float* dt_g = dt + ((size_t)(b * S + t0) * Hh + h) * Dh;
  const float* u_g  = u  + ((size_t)(b * S + t0) * Hh + h) * Dh;
  const float* B_g  = Bm + ((size_t)(b * S + t0) * Hh + h) * Nn;
  const float* C_g  = Cm + ((size_t)(b * S + t0) * Hh + h) * Nn;

  for (int k = d; k < Lc * (Dh / 4); k += Dh) {
    int row = k / (Dh / 4), col4 = (k % (Dh / 4)) * 4;
    async_ld_b128(lds_addr32(&dt_t[row * Dh + col4]), dt_g + (size_t)row * (Hh * Dh) + col4);
    async_ld_b128(lds_addr32(&u_t [row * Dh + col4]), u_g  + (size_t)row * (Hh * Dh) + col4);
  }
  for (int k = d; k < Lc * (Nn / 4); k += Dh) {
    int row = k / (Nn / 4), col4 = (k % (Nn / 4)) * 4;
    async_ld_b128(lds_addr32(&B_t[row * Nn + col4]), B_g + (size_t)row * (Hh * Nn) + col4);
    async_ld_b128(lds_addr32(&C_t[row * Nn + col4]), C_g + (size_t)row * (Hh * Nn) + col4);
  }

  // Overlap per-lane global loads (A row, incoming state, D) with async copies
  float Ar[Nn];
  const float* A_g = A + ((size_t)h * Dh + d) * Nn;
#pragma unroll
  for (int n = 0; n < Nn; ++n) Ar[n] = A_g[n];

  float hs[Nn];
  const float* h0p = h0 + (size_t)blk * (Nn * Dh);
#pragma unroll
  for (int n = 0; n < Nn; ++n) hs[n] = h0p[(size_t)n * Dh + d];

  const float Dcoef = Dv[h];

  wait_async_then_barrier();

  for (int t = 0; t < Lc; ++t) {
    const float dtv = dt_t[t * Dh + d];
    const float uv  = u_t[t * Dh + d];
    const float sc  = -dtv * LOG2E;
    const float dtu = dtv * uv;
    float y = 0.f;
#pragma unroll
    for (int n = 0; n < Nn; ++n) {
      const float a = fast_exp2(sc * Ar[n]);
      hs[n] = __builtin_fmaf(a, hs[n], B_t[t * Nn + n] * dtu);
      y = __builtin_fmaf(hs[n], C_t[t * Nn + n], y);
    }
    out[((size_t)(b * S + t0 + t) * Hh + h) * Dh + d] = __builtin_fmaf(Dcoef, uv, y);
  }
}

// ---------------------------------------------------------------------------
extern "C" void kernel_launch(void* const* d_in, const int* in_sizes, int n_in,
                              void* d_out, int out_size, void* d_ws, size_t ws_size,
                              hipStream_t stream) {
  (void)in_sizes; (void)n_in; (void)out_size; (void)ws_size;
  const float* u  = (const float*)d_in[0];
  const float* dt = (const float*)d_in[1];
  const float* A  = (const float*)d_in[2];
  const float* Bm = (const float*)d_in[3];
  const float* Cm = (const float*)d_in[4];
  const float* Dv = (const float*)d_in[5];
  float* out = (float*)d_out;

  // workspace: [agg: Bb*Hh*CH*Nn*Dh float2 (8MB)] [h0: same count float (4MB)]
  const size_t aggElems = (size_t)Bb * Hh * CH * Nn * Dh;
  float2* agg = (float2*)d_ws;
  float*  h0  = (float*)((char*)d_ws + aggElems * sizeof(float2));

  const int nBlocks = Bb * Hh * CH;           // 512 blocks = 1024 wave32s
  ssm_k1_local<<<nBlocks, Dh, 0, stream>>>(u, dt, A, Bm, agg);

  const int scanThreads = Bb * Hh * Nn * Dh;  // 32768
  ssm_k2_scan<<<(scanThreads + 255) / 256, 256, 0, stream>>>(agg, h0);

  ssm_k3_out<<<nBlocks, Dh, 0, stream>>>(u, dt, A, Bm, Cm, Dv, h0, out);
}